// PointNetEncoder_29197187678656
// MI455X (gfx1250) — compile-verified
//
#include <hip/hip_runtime.h>
#include <hip/hip_bf16.h>

#define B_CL 8
#define N_PTS 4096
#define M_PTS 2048
#define K_NN 16
#define Z_DIM 256

typedef float v2f __attribute__((ext_vector_type(2)));
typedef float v8f __attribute__((ext_vector_type(8)));

// Monotone float <-> uint mapping so float max == unsigned max.
__device__ __forceinline__ unsigned fenc(float f) {
    unsigned u = __float_as_uint(f);
    return (u & 0x80000000u) ? ~u : (u | 0x80000000u);
}
__device__ __forceinline__ float fdec(unsigned u) {
    return (u & 0x80000000u) ? __uint_as_float(u ^ 0x80000000u) : __uint_as_float(~u);
}

// Wave-relative LDS byte address of a __shared__ object: low 32 bits of the
// generic pointer (generic = shared_aperture<<32 | lds_offset on AMDGPU).
__device__ __forceinline__ unsigned lds_byte_addr(const void* p) {
    return (unsigned)(unsigned long long)p;
}

// Async global->LDS copy of 16 bytes (GVS mode: SGPR base + 32-bit VGPR offset).
__device__ __forceinline__ void async_copy_b128(unsigned lds_dst, unsigned src_off,
                                                const void* src_base) {
    asm volatile("global_load_async_to_lds_b128 %0, %1, %2 offset:0"
                 :: "v"(lds_dst), "v"(src_off), "s"(src_base)
                 : "memory");
}
__device__ __forceinline__ void async_wait0() {
    asm volatile("s_wait_asynccnt 0" ::: "memory");
}

// ---------------------------------------------------------------------------
// KNN: per wave, 32 rows. Gram matrix via V_WMMA_F32_16X16X4_F32 (K=3 pad 4),
// distances d = |ri|^2 + |cj|^2 - 2*G staged through LDS, per-lane top-16.
// Column tiles staged via GLOBAL_LOAD_ASYNC_TO_LDS_B128.
// Grid: (B, n/256), block 256 (8 waves).
// ---------------------------------------------------------------------------
__global__ __launch_bounds__(256) void knn_kernel(const float* __restrict__ pts,
                                                  int* __restrict__ knn, int n) {
    __shared__ __align__(16) float colP[256 * 3];  // AoS xyz, async-staged
    __shared__ float colN[256];
    __shared__ float dtile[8][32][17];  // per-wave 32x16 staging, padded stride

    const int b = blockIdx.x;
    const int tid = threadIdx.x;
    const int wave = tid >> 5, lane = tid & 31;
    const int half = lane >> 4, l16 = lane & 15;
    const float* P = pts + (size_t)b * n * 3;
    const int r32 = blockIdx.y * 256 + wave * 32;

    // A operands (16x4, K=3 padded): lanes 0-15 carry K0(x)/K1(y), lanes 16-31 K2(z)/K3(0)
    v2f aOp[2];
    float rowN[2][8];
#pragma unroll
    for (int q = 0; q < 2; ++q) {
        int m = r32 + q * 16 + l16;
        float x = P[3 * m + 0], y = P[3 * m + 1], z = P[3 * m + 2];
        aOp[q][0] = half ? z : x;
        aOp[q][1] = half ? 0.0f : y;
#pragma unroll
        for (int i = 0; i < 8; ++i) {  // rows my D registers map to: q*16 + half*8 + i
            int mr = r32 + q * 16 + half * 8 + i;
            float rx = P[3 * mr], ry = P[3 * mr + 1], rz = P[3 * mr + 2];
            rowN[q][i] = rx * rx + ry * ry + rz * rz;
        }
    }

    float kd[K_NN];
    int ki[K_NN];
#pragma unroll
    for (int s = 0; s < K_NN; ++s) { kd[s] = 3.4e38f; ki[s] = 0; }
    float worst = 3.4e38f;
    int wslot = 0;

    const unsigned colPAddr = lds_byte_addr(colP);

    for (int cb = 0; cb < n; cb += 256) {
        __syncthreads();  // previous chunk fully consumed by all waves
        if (tid < 192) {  // 256 pts * 12B = 3072B = 192 x b128 async transfers
            async_copy_b128(colPAddr + (unsigned)tid * 16u,
                            (unsigned)(cb * 12) + (unsigned)tid * 16u, P);
            async_wait0();
        }
        __syncthreads();
        {
            float x = colP[3 * tid], y = colP[3 * tid + 1], z = colP[3 * tid + 2];
            colN[tid] = x * x + y * y + z * z;
        }
        __syncthreads();
#pragma unroll 1
        for (int sub = 0; sub < 16; ++sub) {
            int cl = sub * 16 + l16;
            float cx = colP[3 * cl], cy = colP[3 * cl + 1], cz = colP[3 * cl + 2];
            float cn = colN[cl];
            v2f bOp;  // B 4x16 mirrored layout
            bOp[0] = half ? cz : cx;
            bOp[1] = half ? 0.0f : cy;
#pragma unroll
            for (int q = 0; q < 2; ++q) {
                v8f acc = {0.f, 0.f, 0.f, 0.f, 0.f, 0.f, 0.f, 0.f};
                acc = __builtin_amdgcn_wmma_f32_16x16x4_f32(
                    false, aOp[q], false, bOp, (short)0, acc, false, false);
#pragma unroll
                for (int i = 0; i < 8; ++i) {
                    float dist = fmaf(-2.0f, acc[i], rowN[q][i] + cn);
                    dtile[wave][q * 16 + half * 8 + i][l16] = dist;
                }
            }
            asm volatile("s_wait_dscnt 0" ::: "memory");  // wave-local LDS RAW
            const int colBase = cb + sub * 16;
#pragma unroll
            for (int c = 0; c < 16; ++c) {
                float d = dtile[wave][lane][c];
                if (d < worst) {
#pragma unroll
                    for (int s = 0; s < K_NN; ++s)
                        if (s == wslot) { kd[s] = d; ki[s] = colBase + c; }
                    float w = kd[0]; int ws_ = 0;
#pragma unroll
                    for (int s = 1; s < K_NN; ++s)
                        if (kd[s] > w) { w = kd[s]; ws_ = s; }
                    worst = w; wslot = ws_;
                }
            }
        }
    }
    int row = r32 + lane;
    int* o = knn + ((size_t)b * n + row) * K_NN;
#pragma unroll
    for (int s = 0; s < K_NN; ++s) o[s] = ki[s];
}

// ---------------------------------------------------------------------------
// Farthest point sampling: one WG per cloud, cloud resident in LDS (async-
// staged AoS), 16 points per thread in registers, wave32 shuffle argmax.
// Word-stride-3 LDS reads are bank-conflict-free (gcd(3,64)=1).
// ---------------------------------------------------------------------------
__global__ __launch_bounds__(256) void fps_kernel(const float* __restrict__ pos,
                                                  int* __restrict__ sel,
                                                  float* __restrict__ p2) {
    __shared__ __align__(16) float sp[N_PTS * 3];  // 48KB AoS cloud
    __shared__ float wval[8];
    __shared__ int widx[8];
    __shared__ int curSh;
    const int b = blockIdx.x, t = threadIdx.x;
    const float* P = pos + (size_t)b * N_PTS * 3;

    {   // 49152B = 3072 x b128; 12 per thread
        const unsigned spAddr = lds_byte_addr(sp);
#pragma unroll
        for (int j = 0; j < 12; ++j) {
            unsigned off = (unsigned)(t * 192 + j * 16);
            async_copy_b128(spAddr + off, off, P);
        }
        async_wait0();
    }
    __syncthreads();

    float px[16], py[16], pz[16], dmin[16];
#pragma unroll
    for (int q = 0; q < 16; ++q) {
        int i = t + q * 256;
        px[q] = sp[3 * i]; py[q] = sp[3 * i + 1]; pz[q] = sp[3 * i + 2];
        dmin[q] = 3.4e38f;
    }
    if (t == 0) {
        sel[(size_t)b * M_PTS] = 0;
        float* pp = p2 + (size_t)b * M_PTS * 3;
        pp[0] = sp[0]; pp[1] = sp[1]; pp[2] = sp[2];
    }
    float cx = sp[0], cy = sp[1], cz = sp[2];
    for (int s = 1; s < M_PTS; ++s) {
        float lv = -3.4e38f;
        int li = 0x7fffffff;
#pragma unroll
        for (int q = 0; q < 16; ++q) {
            float dx = px[q] - cx, dy = py[q] - cy, dz = pz[q] - cz;
            float d = fmaf(dx, dx, fmaf(dy, dy, dz * dz));
            float dm = fminf(dmin[q], d);
            dmin[q] = dm;
            if (dm > lv) { lv = dm; li = t + q * 256; }  // first-max tiebreak
        }
#pragma unroll
        for (int off = 16; off > 0; off >>= 1) {  // wave32 argmax
            float ov = __shfl_xor(lv, off);
            int oi = __shfl_xor(li, off);
            if (ov > lv || (ov == lv && oi < li)) { lv = ov; li = oi; }
        }
        if ((t & 31) == 0) { wval[t >> 5] = lv; widx[t >> 5] = li; }
        __syncthreads();
        if (t == 0) {
            float bv = wval[0]; int bi = widx[0];
#pragma unroll
            for (int w = 1; w < 8; ++w)
                if (wval[w] > bv || (wval[w] == bv && widx[w] < bi)) { bv = wval[w]; bi = widx[w]; }
            curSh = bi;
            sel[(size_t)b * M_PTS + s] = bi;
            float* pp = p2 + ((size_t)b * M_PTS + s) * 3;
            pp[0] = sp[3 * bi]; pp[1] = sp[3 * bi + 1]; pp[2] = sp[3 * bi + 2];
        }
        __syncthreads();
        int cur = curSh;
        cx = sp[3 * cur]; cy = sp[3 * cur + 1]; cz = sp[3 * cur + 2];
    }
}

// ---------------------------------------------------------------------------
// Conv1: msg = [p_j, p_j - p_i] @ W1 + b1, max over k, SiLU. One point/thread.
// ---------------------------------------------------------------------------
__global__ __launch_bounds__(256) void conv1_kernel(const float* __restrict__ pos,
                                                    const int* __restrict__ idx1,
                                                    const float* __restrict__ W1,
                                                    const float* __restrict__ b1,
                                                    float* __restrict__ h) {
    __shared__ float sW[6][32];
    __shared__ float sb[32];
    int t = threadIdx.x;
    if (t < 192) sW[t >> 5][t & 31] = W1[t];
    if (t < 32) sb[t] = b1[t];
    __syncthreads();
    int gid = blockIdx.x * 256 + t;
    int b = gid >> 12;           // /4096
    int i = gid & (N_PTS - 1);
    const float* P = pos + (size_t)b * N_PTS * 3;
    float pix = P[3 * i], piy = P[3 * i + 1], piz = P[3 * i + 2];
    float acc[32];
#pragma unroll
    for (int o = 0; o < 32; ++o) acc[o] = -3.4e38f;
    const int* nb = idx1 + (size_t)gid * K_NN;
    for (int k = 0; k < K_NN; ++k) {
        int j = nb[k];
        float xj = P[3 * j], yj = P[3 * j + 1], zj = P[3 * j + 2];
        float rx = xj - pix, ry = yj - piy, rz = zj - piz;
#pragma unroll
        for (int o = 0; o < 32; ++o) {
            float m = sb[o];
            m = fmaf(xj, sW[0][o], m); m = fmaf(yj, sW[1][o], m); m = fmaf(zj, sW[2][o], m);
            m = fmaf(rx, sW[3][o], m); m = fmaf(ry, sW[4][o], m); m = fmaf(rz, sW[5][o], m);
            acc[o] = fmaxf(acc[o], m);
        }
    }
    float* ho = h + (size_t)gid * 32;
#pragma unroll
    for (int o = 0; o < 32; ++o) {
        float a = acc[o];
        ho[o] = a * (1.0f / (1.0f + __expf(-a)));
    }
}

// ---------------------------------------------------------------------------
// Conv2: msg = [h_j(32), rel(3)] @ W2 + b2, max over k, SiLU, then global max
// into g via monotone-encoded atomicMax.
// ---------------------------------------------------------------------------
__global__ __launch_bounds__(256) void conv2_kernel(const float* __restrict__ p2,
                                                    const int* __restrict__ idx2,
                                                    const int* __restrict__ sel,
                                                    const float* __restrict__ h,
                                                    const float* __restrict__ W2,
                                                    const float* __restrict__ b2,
                                                    unsigned* __restrict__ g_u) {
    __shared__ float sW[35][32];
    __shared__ float sb[32];
    __shared__ float pmax[8][32];
    int t = threadIdx.x;
    for (int i = t; i < 35 * 32; i += 256) sW[i >> 5][i & 31] = W2[i];
    if (t < 32) sb[t] = b2[t];
    __syncthreads();
    int gid = blockIdx.x * 256 + t;
    int b = gid >> 11;            // /2048
    int m = gid & (M_PTS - 1);
    const float* P2 = p2 + (size_t)b * M_PTS * 3;
    const int* selb = sel + (size_t)b * M_PTS;
    const float* H = h + (size_t)b * N_PTS * 32;
    float pix = P2[3 * m], piy = P2[3 * m + 1], piz = P2[3 * m + 2];
    float acc[32];
#pragma unroll
    for (int o = 0; o < 32; ++o) acc[o] = -3.4e38f;
    const int* nb = idx2 + (size_t)gid * K_NN;
    for (int k = 0; k < K_NN; ++k) {
        int jm = nb[k];
        float xj = P2[3 * jm], yj = P2[3 * jm + 1], zj = P2[3 * jm + 2];
        float rx = xj - pix, ry = yj - piy, rz = zj - piz;
        const float* fj = H + (size_t)selb[jm] * 32;
        float msg[32];
#pragma unroll
        for (int o = 0; o < 32; ++o) {
            float v = sb[o];
            v = fmaf(rx, sW[32][o], v); v = fmaf(ry, sW[33][o], v); v = fmaf(rz, sW[34][o], v);
            msg[o] = v;
        }
#pragma unroll 4
        for (int f = 0; f < 32; ++f) {
            float xf = fj[f];
#pragma unroll
            for (int o = 0; o < 32; ++o) msg[o] = fmaf(xf, sW[f][o], msg[o]);
        }
#pragma unroll
        for (int o = 0; o < 32; ++o) acc[o] = fmaxf(acc[o], msg[o]);
    }
#pragma unroll
    for (int o = 0; o < 32; ++o) {
        float a = acc[o];
        acc[o] = a * (1.0f / (1.0f + __expf(-a)));
    }
    // wave32 max-reduce each channel, then cross-wave via LDS, 32 atomics/WG
#pragma unroll
    for (int o = 0; o < 32; ++o) {
        float v = acc[o];
#pragma unroll
        for (int off = 16; off > 0; off >>= 1) v = fmaxf(v, __shfl_xor(v, off));
        if ((t & 31) == 0) pmax[t >> 5][o] = v;
    }
    __syncthreads();
    if (t < 32) {
        float v = pmax[0][t];
#pragma unroll
        for (int w = 1; w < 8; ++w) v = fmaxf(v, pmax[w][t]);
        atomicMax(&g_u[b * 32 + t], fenc(v));
    }
}

__global__ void init_g_kernel(unsigned* __restrict__ g_u) {
    g_u[threadIdx.x] = 0u;  // encode-space minimum
}

__global__ __launch_bounds__(256) void final_kernel(const unsigned* __restrict__ g_u,
                                                    const float* __restrict__ W3,
                                                    const float* __restrict__ b3,
                                                    float* __restrict__ out) {
    __shared__ float gv[32];
    int bcl = blockIdx.x, t = threadIdx.x;
    if (t < 32) gv[t] = fdec(g_u[bcl * 32 + t]);
    __syncthreads();
    float s = b3[t];
#pragma unroll 8
    for (int f = 0; f < 32; ++f) s = fmaf(gv[f], W3[f * Z_DIM + t], s);
    out[bcl * Z_DIM + t] = s;
}

extern "C" void kernel_launch(void* const* d_in, const int* in_sizes, int n_in,
                              void* d_out, int out_size, void* d_ws, size_t ws_size,
                              hipStream_t stream) {
    (void)in_sizes; (void)n_in; (void)out_size; (void)ws_size;
    const float* pos = (const float*)d_in[0];
    // d_in[1] = batch ids (equal-size sorted clouds) -- implicit in layout
    const float* W1 = (const float*)d_in[2];
    const float* b1 = (const float*)d_in[3];
    const float* W2 = (const float*)d_in[4];
    const float* b2 = (const float*)d_in[5];
    const float* W3 = (const float*)d_in[6];
    const float* b3 = (const float*)d_in[7];
    float* out = (float*)d_out;

    char* ws = (char*)d_ws;
    size_t o = 0;
    int* idx1 = (int*)(ws + o);      o += (size_t)B_CL * N_PTS * K_NN * sizeof(int);
    float* h = (float*)(ws + o);     o += (size_t)B_CL * N_PTS * 32 * sizeof(float);
    int* sel = (int*)(ws + o);       o += (size_t)B_CL * M_PTS * sizeof(int);
    float* p2 = (float*)(ws + o);    o += (size_t)B_CL * M_PTS * 3 * sizeof(float);
    int* idx2 = (int*)(ws + o);      o += (size_t)B_CL * M_PTS * K_NN * sizeof(int);
    unsigned* g_u = (unsigned*)(ws + o);

    fps_kernel<<<B_CL, 256, 0, stream>>>(pos, sel, p2);
    knn_kernel<<<dim3(B_CL, N_PTS / 256), 256, 0, stream>>>(pos, idx1, N_PTS);
    conv1_kernel<<<B_CL * N_PTS / 256, 256, 0, stream>>>(pos, idx1, W1, b1, h);
    knn_kernel<<<dim3(B_CL, M_PTS / 256), 256, 0, stream>>>(p2, idx2, M_PTS);
    init_g_kernel<<<1, 256, 0, stream>>>(g_u);
    conv2_kernel<<<B_CL * M_PTS / 256, 256, 0, stream>>>(p2, idx2, sel, h, W2, b2, g_u);
    final_kernel<<<B_CL, 256, 0, stream>>>(g_u, W3, b3, out);
}